// RGTLayer_18013092839759
// MI455X (gfx1250) — compile-verified
//
#include <hip/hip_runtime.h>
#include <hip/hip_bf16.h>

// ---------------- problem constants (from reference) ----------------
#define NN   50000      // nodes
#define INC  128        // in_channels == out_dim
#define HH   4          // heads
#define DD   128        // per-head dim
#define RR   3          // relations
#define EE   400000     // edges per relation
#define HD   (HH*DD)    // 512
#define LDST 132        // LDS row stride (pad: 16B-aligned rows + conflict-free b64 reads)

typedef float v2f __attribute__((ext_vector_type(2)));
typedef float v8f __attribute__((ext_vector_type(8)));

// D = A(16x4) * B(4x16) + C, fp32 WMMA (CDNA5)
__device__ __forceinline__ v8f wmma4(v2f a, v2f b, v8f c) {
  return __builtin_amdgcn_wmma_f32_16x16x4_f32(false, a, false, b, (short)0, c, false, false);
}

__device__ __forceinline__ void atomicMaxFloat(float* addr, float val) {
  if (val >= 0.f) atomicMax((int*)addr, __float_as_int(val));
  else            atomicMin((unsigned int*)addr, __float_as_uint(val));
}

// Stage a 16x128 fp32 tile (row stride INC) into LDS (row stride LDST),
// cooperatively across the block (b128 global load + b128 LDS store).
__device__ __forceinline__ void stage_tile16(const float* __restrict__ src,
                                             float* __restrict__ dstLds) {
  for (int t = threadIdx.x; t < 512; t += blockDim.x) {
    const int rr = t >> 5;            // row 0..15
    const int c4 = (t & 31) << 2;     // float4 column offset
    const float4 v = *(const float4*)(src + rr * INC + c4);
    *(float4*)(dstLds + rr * LDST + c4) = v;
  }
}

// ---------------- fills ----------------
__global__ void fill_f32(float* __restrict__ p, float v, long n) {
  long i = (long)blockIdx.x * blockDim.x + threadIdx.x;
  if (i < n) p[i] = v;
}

// ---------------- K1: Q,K,V,S projections via fp32 WMMA ----------------
// 4 waves/block share one LDS-staged A tile; each wave owns a 16x16 tile.
// 104 column tiles: [0,512)=Q, [512,1024)=K, [1024,1536)=V, [1536,1664)=S
__global__ __launch_bounds__(128) void qkvs_gemm(
    const float* __restrict__ x,
    const float* __restrict__ Wq, const float* __restrict__ bq,
    const float* __restrict__ Wk, const float* __restrict__ bk,
    const float* __restrict__ Wv, const float* __restrict__ bv,
    const float* __restrict__ Ws, const float* __restrict__ bs,
    float* __restrict__ Q, float* __restrict__ Kb,
    float* __restrict__ V, float* __restrict__ S) {
  __shared__ float As[16 * LDST];
  const int wave = threadIdx.x >> 5;
  const int lane = threadIdx.x & 31;
  const int half = lane >> 4, l = lane & 15;
  const int ct = blockIdx.x * 4 + wave;       // 0..103
  const int row0 = blockIdx.y * 16;

  stage_tile16(x + (size_t)row0 * INC, As);

  const int c = ct * 16;
  const float* W; const float* bias; float* Out; int ldb; int col0;
  if (c < 512)       { W = Wq; bias = bq; Out = Q;  ldb = 512; col0 = c; }
  else if (c < 1024) { W = Wk; bias = bk; Out = Kb; ldb = 512; col0 = c - 512; }
  else if (c < 1536) { W = Wv; bias = bv; Out = V;  ldb = 512; col0 = c - 1024; }
  else               { W = Ws; bias = bs; Out = S;  ldb = 128; col0 = c - 1536; }

  __syncthreads();

  v8f acc = {};
  const float* ArowL = As + l * LDST;
  for (int k = 0; k < INC; k += 4) {
    const int ka = k + 2 * half;
    v2f a, b;
    a.x = ArowL[ka]; a.y = ArowL[ka + 1];
    b.x = W[(size_t)ka * ldb + col0 + l];
    b.y = W[(size_t)(ka + 1) * ldb + col0 + l];
    acc = wmma4(a, b, acc);
  }
  const float bb = bias[col0 + l];
#pragma unroll
  for (int j = 0; j < 8; ++j) {
    const int m = row0 + j + 8 * half;
    Out[(size_t)m * ldb + col0 + l] = acc[j] + bb;
  }
}

// ---------------- K2: per-edge logits (4 heads) + segment max ----------------
__global__ __launch_bounds__(256) void edge_logits_max(
    const float* __restrict__ Q, const float* __restrict__ Kb,
    const int* __restrict__ src, const int* __restrict__ dst,
    float* __restrict__ logits, float* __restrict__ mmax) {
  const int wave = threadIdx.x >> 5, lane = threadIdx.x & 31;
  const long e = (long)blockIdx.x * 8 + wave;
  if (e >= EE) return;
  const int s = src[e], d = dst[e];
  const float4* q4 = (const float4*)(Q + (size_t)d * HD);
  const float4* k4 = (const float4*)(Kb + (size_t)s * HD);
  float part = 0.f;
#pragma unroll
  for (int i = 0; i < 4; ++i) {
    float4 qa = q4[lane * 4 + i];
    float4 ka = k4[lane * 4 + i];
    part += qa.x * ka.x + qa.y * ka.y + qa.z * ka.z + qa.w * ka.w;
  }
  part += __shfl_xor(part, 1, 32);
  part += __shfl_xor(part, 2, 32);
  part += __shfl_xor(part, 4, 32);
  if ((lane & 7) == 0) {
    const int h = lane >> 3;
    const float lg = part * 0.0883883476483184f;  // 1/sqrt(128)
    logits[e * HH + h] = lg;
    atomicMaxFloat(&mmax[(size_t)d * HH + h], lg);
  }
}

// ---------------- K3: ex = exp(logit - m[dst]); den[dst] += ex ----------------
__global__ __launch_bounds__(256) void edge_exp_den(
    const int* __restrict__ dst, float* __restrict__ logits,
    const float* __restrict__ mmax, float* __restrict__ den) {
  const long i = (long)blockIdx.x * blockDim.x + threadIdx.x;
  if (i >= (long)EE * HH) return;
  const long e = i >> 2; const int h = (int)(i & 3);
  const int d = dst[e];
  const float ex = __expf(logits[i] - mmax[(size_t)d * HH + h]);
  logits[i] = ex;
  atomicAdd(&den[(size_t)d * HH + h], ex);
}

// ---------------- K4: msg[dst] += alpha * v[src] ----------------
__global__ __launch_bounds__(256) void edge_aggregate(
    const float* __restrict__ V, const int* __restrict__ src,
    const int* __restrict__ dst, const float* __restrict__ ex,
    const float* __restrict__ den, float* __restrict__ msg) {
  const int wave = threadIdx.x >> 5, lane = threadIdx.x & 31;
  const long e = (long)blockIdx.x * 8 + wave;
  if (e >= EE) return;
  const int s = src[e], d = dst[e];
  const int h = lane >> 3;
  const float alpha = ex[e * HH + h] / den[(size_t)d * HH + h];
  const float* vp = V + (size_t)s * HD + lane * 16;
  float* mp = msg + (size_t)d * HD + lane * 16;
#pragma unroll
  for (int i = 0; i < 16; ++i) atomicAdd(&mp[i], alpha * vp[i]);
}

// ---------------- K5: U = mean_h(msg) + S  (in place over S) ----------------
__global__ __launch_bounds__(256) void finalize_U(
    const float* __restrict__ msg, float* __restrict__ S) {
  const long i = (long)blockIdx.x * blockDim.x + threadIdx.x;
  if (i >= (long)NN * INC) return;
  const long n = i >> 7; const int c = (int)(i & 127);
  const float* m = msg + n * HD + c;
  S[i] += 0.25f * (m[0] + m[128] + m[256] + m[384]);
}

// ---------------- K6: gating GEMM (WMMA, K=256) + gate + write Hst ----------------
// 4 waves/block share LDS-staged U and x tiles (16x128 each).
__global__ __launch_bounds__(128) void gate_hst(
    const float* __restrict__ U, const float* __restrict__ x,
    const float* __restrict__ Wg, const float* __restrict__ bg,
    float* __restrict__ Hst, int r) {
  __shared__ float Us[16 * LDST];
  __shared__ float Xs[16 * LDST];
  const int wave = threadIdx.x >> 5, lane = threadIdx.x & 31;
  const int half = lane >> 4, l = lane & 15;
  const int col0 = (blockIdx.x * 4 + wave) * 16;  // 0..112
  const int row0 = blockIdx.y * 16;

  stage_tile16(U + (size_t)row0 * INC, Us);
  stage_tile16(x + (size_t)row0 * INC, Xs);
  __syncthreads();

  v8f acc = {};
  const float* UrowL = Us + l * LDST;
  const float* XrowL = Xs + l * LDST;
  for (int k = 0; k < INC; k += 4) {           // U part (Wg rows 0..127)
    const int ka = k + 2 * half;
    v2f a, b;
    a.x = UrowL[ka]; a.y = UrowL[ka + 1];
    b.x = Wg[(size_t)ka * INC + col0 + l];
    b.y = Wg[(size_t)(ka + 1) * INC + col0 + l];
    acc = wmma4(a, b, acc);
  }
  for (int k = 0; k < INC; k += 4) {           // x part (Wg rows 128..255)
    const int ka = k + 2 * half;
    v2f a, b;
    a.x = XrowL[ka]; a.y = XrowL[ka + 1];
    b.x = Wg[(size_t)(INC + ka) * INC + col0 + l];
    b.y = Wg[(size_t)(INC + ka + 1) * INC + col0 + l];
    acc = wmma4(a, b, acc);
  }
  const float bb = bg[col0 + l];
#pragma unroll
  for (int j = 0; j < 8; ++j) {
    const int m = row0 + j + 8 * half;
    const int rloc = j + 8 * half;
    const float u  = Us[rloc * LDST + col0 + l];
    const float xv = Xs[rloc * LDST + col0 + l];
    const float z  = 1.f / (1.f + __expf(-(acc[j] + bb)));
    Hst[(size_t)m * (RR * INC) + r * INC + col0 + l] = tanhf(u) * z + xv * (1.f - z);
  }
}

// ---------------- K7: semantic head MLP + score row-sums ----------------
// One wave per block; stage the 16-row Hst tile once, reuse across 8 col tiles.
__global__ __launch_bounds__(32) void semantic_scores(
    const float* __restrict__ Hst, const float* __restrict__ W1,
    const float* __restrict__ b1, const float* __restrict__ w2,
    float* __restrict__ score_sum) {
  __shared__ float As[16 * LDST];
  const int lane = threadIdx.x & 31;
  const int half = lane >> 4, l = lane & 15;
  const int h = blockIdx.y;
  const int row0 = blockIdx.x * 16;
  const float* W   = W1 + (size_t)h * INC * INC;
  const float* bia = b1 + h * INC;
  const float* w2h = w2 + h * INC;

  stage_tile16(Hst + (size_t)row0 * INC, As);
  __syncthreads();

  const float* ArowL = As + l * LDST;
  float s[8] = {0.f, 0.f, 0.f, 0.f, 0.f, 0.f, 0.f, 0.f};
  for (int ctile = 0; ctile < 8; ++ctile) {
    const int col0 = ctile * 16;
    v8f acc = {};
    for (int k = 0; k < INC; k += 4) {
      const int ka = k + 2 * half;
      v2f a, b;
      a.x = ArowL[ka]; a.y = ArowL[ka + 1];
      b.x = W[(size_t)ka * INC + col0 + l];
      b.y = W[(size_t)(ka + 1) * INC + col0 + l];
      acc = wmma4(a, b, acc);
    }
    const float bb = bia[col0 + l];
    const float ww = w2h[col0 + l];
#pragma unroll
    for (int j = 0; j < 8; ++j) s[j] += tanhf(acc[j] + bb) * ww;
  }
#pragma unroll
  for (int j = 0; j < 8; ++j) {     // reduce across 16 lanes of each half
    s[j] += __shfl_xor(s[j], 1, 32);
    s[j] += __shfl_xor(s[j], 2, 32);
    s[j] += __shfl_xor(s[j], 4, 32);
    s[j] += __shfl_xor(s[j], 8, 32);
  }
  if (l == 0) {
    float a3[3] = {0.f, 0.f, 0.f};
#pragma unroll
    for (int j = 0; j < 8; ++j) a3[(row0 + j + 8 * half) % RR] += s[j];
    for (int rr = 0; rr < RR; ++rr) atomicAdd(&score_sum[h * RR + rr], a3[rr]);
  }
}

// ---------------- K8: beta softmax over relations -> combine coeffs ----------------
__global__ void beta_coeff(const float* __restrict__ score_sum, float* __restrict__ coeff) {
  if (threadIdx.x == 0 && blockIdx.x == 0) {
    float c[RR] = {0.f, 0.f, 0.f};
    for (int h = 0; h < HH; ++h) {
      float sm[RR], mx = -1e30f;
      for (int r = 0; r < RR; ++r) {
        sm[r] = score_sum[h * RR + r] * (1.f / (float)NN);
        if (sm[r] > mx) mx = sm[r];
      }
      float sum = 0.f;
      for (int r = 0; r < RR; ++r) { sm[r] = __expf(sm[r] - mx); sum += sm[r]; }
      for (int r = 0; r < RR; ++r) c[r] += sm[r] / sum;
    }
    for (int r = 0; r < RR; ++r) coeff[r] = c[r] * (1.f / (float)HH);
  }
}

// ---------------- K9: out[n,d] = sum_r coeff[r] * Hst[n,r,d] ----------------
__global__ __launch_bounds__(256) void final_out(
    const float* __restrict__ Hst, const float* __restrict__ coeff,
    float* __restrict__ out) {
  const long i = (long)blockIdx.x * blockDim.x + threadIdx.x;
  if (i >= (long)NN * INC) return;
  const long n = i >> 7; const int c = (int)(i & 127);
  const float* hp = Hst + n * (RR * INC) + c;
  out[i] = coeff[0] * hp[0] + coeff[1] * hp[INC] + coeff[2] * hp[2 * INC];
}

// ---------------- launch ----------------
extern "C" void kernel_launch(void* const* d_in, const int* in_sizes, int n_in,
                              void* d_out, int out_size, void* d_ws, size_t ws_size,
                              hipStream_t stream) {
  const float* x    = (const float*)d_in[0];
  const int*   eidx = (const int*)d_in[1];          // [R,2,E]
  const float* Wq = (const float*)d_in[2];
  const float* bq = (const float*)d_in[3];
  const float* Wk = (const float*)d_in[4];
  const float* bk = (const float*)d_in[5];
  const float* Wv = (const float*)d_in[6];
  const float* bv = (const float*)d_in[7];
  const float* Ws = (const float*)d_in[8];
  const float* bs = (const float*)d_in[9];
  const float* Wg = (const float*)d_in[10];
  const float* bg = (const float*)d_in[11];
  const float* W1 = (const float*)d_in[12];
  const float* b1 = (const float*)d_in[13];
  const float* w2 = (const float*)d_in[14];
  float* out = (float*)d_out;

  // workspace carve-up (floats)
  float* ws = (float*)d_ws;
  size_t off = 0;
  float* Q      = ws + off; off += (size_t)NN * HD;
  float* Kb     = ws + off; off += (size_t)NN * HD;
  float* V      = ws + off; off += (size_t)NN * HD;
  float* S      = ws + off; off += (size_t)NN * INC;   // becomes U in place
  float* logits = ws + off; off += (size_t)EE * HH;    // becomes ex in place
  float* mmax   = ws + off; off += (size_t)NN * HH;
  float* den    = ws + off; off += (size_t)NN * HH;
  float* msg    = ws + off; off += (size_t)NN * HD;
  float* Hst    = ws + off; off += (size_t)NN * RR * INC;
  float* score_sum = ws + off; off += 16;
  float* coeff     = ws + off; off += 16;

  const dim3 blk256(256), blk128(128), blk32(32);
  const int rowTiles = NN / 16;             // 3125

  for (int r = 0; r < RR; ++r) {
    // reset per-relation accumulators
    fill_f32<<<dim3(((long)NN * HH + 255) / 256), blk256, 0, stream>>>(mmax, -1e30f, (long)NN * HH);
    fill_f32<<<dim3(((long)NN * HH + 255) / 256), blk256, 0, stream>>>(den, 0.f, (long)NN * HH);
    fill_f32<<<dim3(((long)NN * HD + 255) / 256), blk256, 0, stream>>>(msg, 0.f, (long)NN * HD);

    qkvs_gemm<<<dim3(26, rowTiles), blk128, 0, stream>>>(
        x,
        Wq + (size_t)r * INC * HD, bq + (size_t)r * HD,
        Wk + (size_t)r * INC * HD, bk + (size_t)r * HD,
        Wv + (size_t)r * INC * HD, bv + (size_t)r * HD,
        Ws + (size_t)r * INC * INC, bs + (size_t)r * INC,
        Q, Kb, V, S);

    const int* src = eidx + (size_t)r * 2 * EE;
    const int* dst = src + EE;

    edge_logits_max<<<dim3(EE / 8), blk256, 0, stream>>>(Q, Kb, src, dst, logits, mmax);
    edge_exp_den<<<dim3(((long)EE * HH + 255) / 256), blk256, 0, stream>>>(dst, logits, mmax, den);
    edge_aggregate<<<dim3(EE / 8), blk256, 0, stream>>>(V, src, dst, logits, den, msg);
    finalize_U<<<dim3(((long)NN * INC + 255) / 256), blk256, 0, stream>>>(msg, S);
    gate_hst<<<dim3(2, rowTiles), blk128, 0, stream>>>(S, x, Wg, bg, Hst, r);
  }

  fill_f32<<<dim3(1), blk32, 0, stream>>>(score_sum, 0.f, 16);
  semantic_scores<<<dim3((NN * RR) / 16, HH), blk32, 0, stream>>>(Hst, W1, b1, w2, score_sum);
  beta_coeff<<<dim3(1), blk32, 0, stream>>>(score_sum, coeff);
  final_out<<<dim3(((long)NN * INC + 255) / 256), blk256, 0, stream>>>(Hst, coeff, out);
}